// TaskAlignedAssigner_33466385170495
// MI455X (gfx1250) — compile-verified
//
#include <hip/hip_runtime.h>
#include <stdint.h>

// ---------------------------------------------------------------------------
// TaskAlignedAssigner for MI455X (gfx1250, wave32).
// B=16, L=8400, C=80, N=64, TOPK=13, ALPHA=1, BETA=6.
// No GEMM content -> no WMMA. CDNA5 paths used: async global->LDS staging
// (ASYNCcnt + s_wait_asynccnt), non-temporal stores for the 80MB write-once
// output, global_prefetch. Assumes bg_index == C (true for this config).
// ---------------------------------------------------------------------------

#define B_     16
#define L_     8400
#define C_     80
#define N_     64
#define TOPK_  13
#define EPS_F  1e-9f

typedef float v4f __attribute__((ext_vector_type(4)));
typedef int   v4i __attribute__((vector_size(16)));   // matches builtin proto

#if __has_builtin(__builtin_amdgcn_global_load_async_to_lds_b128) && \
    __has_builtin(__builtin_amdgcn_s_wait_asynccnt)
#define USE_ASYNC_LDS 1
#else
#define USE_ASYNC_LDS 0
#endif

// ---------------------------------------------------------------------------
// Kernel 2 (defined first so the disasm snippet shows the async-LDS path):
// per anchor (b,l): resolve multiple-gt anchors via argmax-iou, emit
// labels/bboxes/mask_positive, record (n*, metric*), atomicMax per-gt maxima
// of masked metric & iou. GT tables staged via async global->LDS.
// ---------------------------------------------------------------------------
__global__ __launch_bounds__(256) void TAA_resolve(
    const float* __restrict__ pred_scores,   // (B,L,C)
    const float* __restrict__ pred_bboxes,   // (B,L,4)
    const int*   __restrict__ gt_labels,     // (B,N)
    const float* __restrict__ gt_bboxes,     // (B,N,4)
    const int*   __restrict__ bg_index_p,
    const unsigned int* __restrict__ maskA,  // (B,L,2)
    unsigned int* __restrict__ maxMet,       // (B,N) f32 bits, init 0
    unsigned int* __restrict__ maxIou,       // (B,N) f32 bits, init 0
    int*   __restrict__ rec_n,               // (B,L)
    float* __restrict__ rec_m,               // (B,L)
    float* __restrict__ out_labels,          // (B,L)
    float* __restrict__ out_bboxes,          // (B,L,4)
    float* __restrict__ out_maskpos)         // (B,N,L)
{
  __shared__ v4f sgt[N_];
  __shared__ __align__(16) int slbl[N_];

  const int b   = blockIdx.y;
  const int tid = threadIdx.x;
  const int l   = blockIdx.x * 256 + tid;

#if USE_ASYNC_LDS
  // CDNA5 async global->LDS copy (tracked with ASYNCcnt).
  // Builtin proto (probe-confirmed via diagnostics): (v4i as1*, v4i as3*, imm, imm)
  if (tid < N_) {
    __builtin_amdgcn_global_load_async_to_lds_b128(
        (__attribute__((address_space(1))) v4i*)
            (gt_bboxes + (size_t)(b * N_ + tid) * 4),
        (__attribute__((address_space(3))) v4i*)(&sgt[tid]), 0, 0);
  }
  if (tid < N_ / 4) {
    __builtin_amdgcn_global_load_async_to_lds_b128(
        (__attribute__((address_space(1))) v4i*)
            (gt_labels + b * N_ + tid * 4),
        (__attribute__((address_space(3))) v4i*)(&slbl[tid * 4]), 0, 0);
  }
  __builtin_amdgcn_s_wait_asynccnt(0);
  asm volatile("" ::: "memory");   // keep LDS reads below the wait
  __syncthreads();
#else
  if (tid < N_) {
    sgt[tid]  = ((const v4f*)gt_bboxes)[b * N_ + tid];
    slbl[tid] = gt_labels[b * N_ + tid];
  }
  __syncthreads();
#endif

  const bool  active = (l < L_);
  const int   lc     = active ? l : (L_ - 1);
  const size_t ml    = (size_t)b * L_ + lc;
  const int   bg     = *bg_index_p;

  v4f   p     = ((const v4f*)pred_bboxes)[ml];
  float areap = (p.z - p.x) * (p.w - p.y);

  unsigned int w0 = maskA[ml * 2 + 0];
  unsigned int w1 = maskA[ml * 2 + 1];
  int psum = __popc(w0) + __popc(w1);          // mask_positive_sum (prelim)

  // argmax_n iou (first occurrence), over ALL gts, as in reference
  float best = -1.0f; int bestn = 0;
  for (int n = 0; n < N_; ++n) {
    v4f g = sgt[n];
    float iw    = fmaxf(fminf(g.z, p.z) - fmaxf(g.x, p.x), 0.0f);
    float ih    = fmaxf(fminf(g.w, p.w) - fmaxf(g.y, p.y), 0.0f);
    float inter = iw * ih;
    float areag = (g.z - g.x) * (g.w - g.y);
    float iou   = inter / (areag + areap - inter + EPS_F);
    if (iou > best) { best = iou; bestn = n; }
  }

  unsigned int f0 = w0, f1 = w1;
  if (psum > 1) {                              // replace with is_max_iou
    f0 = (bestn < 32) ? (1u << bestn) : 0u;
    f1 = (bestn >= 32) ? (1u << (bestn - 32)) : 0u;
  }
  int nstar = -1;                              // final mask has <= 1 bit
  if (f0)      nstar = __ffs(f0) - 1;
  else if (f1) nstar = 32 + __ffs(f1) - 1;

  float mstar = 0.0f;
  if (nstar >= 0) {
    v4f g = sgt[nstar];
    float iw    = fmaxf(fminf(g.z, p.z) - fmaxf(g.x, p.x), 0.0f);
    float ih    = fmaxf(fminf(g.w, p.w) - fmaxf(g.y, p.y), 0.0f);
    float inter = iw * ih;
    float areag = (g.z - g.x) * (g.w - g.y);
    float iou   = inter / (areag + areap - inter + EPS_F);
    float s     = pred_scores[ml * C_ + slbl[nstar]];
    float iou2  = iou * iou;
    mstar = s * iou2 * iou2 * iou2;            // alignment metric (no in_gts)
    if (active) {                              // values >= 0: uint max == f32 max
      atomicMax(&maxMet[b * N_ + nstar], __float_as_uint(mstar));
      atomicMax(&maxIou[b * N_ + nstar], __float_as_uint(iou));
    }
  }

  if (active) {
    rec_n[ml] = nstar;
    rec_m[ml] = mstar;
    out_labels[ml] = (float)((nstar >= 0) ? slbl[nstar] : bg);
    ((v4f*)out_bboxes)[ml] = sgt[(nstar >= 0) ? nstar : 0];
#pragma unroll 4
    for (int n = 0; n < N_; ++n) {             // coalesced per-n columns
      unsigned int bit = (n < 32) ? ((f0 >> n) & 1u) : ((f1 >> (n - 32)) & 1u);
      __builtin_nontemporal_store((float)bit,
          out_maskpos + (size_t)(b * N_ + n) * L_ + l);
    }
  }
}

// ---------------------------------------------------------------------------
// Kernel 1: per (b,n) compute metric = score(label_n) * iou^6 * in_gts over all
// L anchors, cache in LDS, iteratively extract top-13 (tie-break: smaller
// index, matching jax.lax.top_k stability), set bit n in the per-anchor mask
// when the winner is inside the gt box and the gt is not padded.
// ---------------------------------------------------------------------------
__global__ __launch_bounds__(256) void TAA_topk(
    const float* __restrict__ pred_scores,   // (B,L,C)
    const float* __restrict__ pred_bboxes,   // (B,L,4)
    const float* __restrict__ anchor_points, // (L,2)
    const int*   __restrict__ gt_labels,     // (B,N)
    const float* __restrict__ gt_bboxes,     // (B,N,4)
    const float* __restrict__ pad_gt_mask,   // (B,N)
    unsigned int* __restrict__ maskA)        // (B,L,2) bitmask of positive gts
{
  __shared__ float sm[L_];      // 33.6 KB of metrics
  __shared__ float rv[256];
  __shared__ int   ri[256];

  const int b   = blockIdx.x / N_;
  const int n   = blockIdx.x % N_;
  const int tid = threadIdx.x;

  const float pad   = pad_gt_mask[b * N_ + n];
  const int   lbl   = gt_labels[b * N_ + n];
  const v4f   g     = ((const v4f*)gt_bboxes)[b * N_ + n];
  const float areag = (g.z - g.x) * (g.w - g.y);

  for (int l = tid; l < L_; l += 256) {
    if (l + 256 < L_)
      __builtin_prefetch(&pred_bboxes[((size_t)b * L_ + l + 256) * 4], 0, 1);
    v4f p = ((const v4f*)pred_bboxes)[(size_t)b * L_ + l];
    float iw    = fmaxf(fminf(g.z, p.z) - fmaxf(g.x, p.x), 0.0f);
    float ih    = fmaxf(fminf(g.w, p.w) - fmaxf(g.y, p.y), 0.0f);
    float inter = iw * ih;
    float areap = (p.z - p.x) * (p.w - p.y);
    float iou   = inter / (areag + areap - inter + EPS_F);
    float s     = pred_scores[((size_t)b * L_ + l) * C_ + lbl];
    float iou2  = iou * iou;
    float align = s * iou2 * iou2 * iou2;      // ALPHA=1, BETA=6
    float ax = anchor_points[2 * l], ay = anchor_points[2 * l + 1];
    float delta = fminf(fminf(ax - g.x, ay - g.y), fminf(g.z - ax, g.w - ay));
    sm[l] = (delta > EPS_F) ? align : 0.0f;    // metric * is_in_gts
  }
  __syncthreads();

  for (int k = 0; k < TOPK_; ++k) {
    // local scan (ascending l, strict '>' keeps smallest index on ties)
    float bv = -1.0f; int bi = 0x7fffffff;
    for (int l = tid; l < L_; l += 256) {
      float v = sm[l];
      if (v > bv) { bv = v; bi = l; }
    }
    rv[tid] = bv; ri[tid] = bi;
    __syncthreads();
    for (int s2 = 128; s2 > 0; s2 >>= 1) {
      if (tid < s2) {
        float ov = rv[tid + s2]; int oi = ri[tid + s2];
        if (ov > rv[tid] || (ov == rv[tid] && oi < ri[tid])) {
          rv[tid] = ov; ri[tid] = oi;
        }
      }
      __syncthreads();
    }
    if (tid == 0) {
      int wi = ri[0];
      sm[wi] = -2.0f;                           // exclude from further rounds
      if (pad != 0.0f) {
        float ax = anchor_points[2 * wi], ay = anchor_points[2 * wi + 1];
        float delta = fminf(fminf(ax - g.x, ay - g.y),
                            fminf(g.z - ax, g.w - ay));
        if (delta > EPS_F)                      // is_in_gts filter on topk
          atomicOr(&maskA[((size_t)b * L_ + wi) * 2 + (n >> 5)],
                   1u << (n & 31));
      }
    }
    __syncthreads();
  }
}

// ---------------------------------------------------------------------------
// Kernel 3: assigned_scores = one_hot(label) * m*/(maxMet+eps)*maxIou.
// 43 MB write-once -> non-temporal 128-bit stores.
// ---------------------------------------------------------------------------
__global__ __launch_bounds__(256) void TAA_scores(
    const int*   __restrict__ rec_n,
    const float* __restrict__ rec_m,
    const unsigned int* __restrict__ maxMet,
    const unsigned int* __restrict__ maxIou,
    const int*   __restrict__ gt_labels,
    float* __restrict__ out_scores)            // (B,L,C)
{
  const int idx = blockIdx.x * 256 + threadIdx.x;   // b*L + l
  if (idx >= B_ * L_) return;
  const int b  = idx / L_;
  const int ns = rec_n[idx];
  float am = 0.0f; int lbl = -1;
  if (ns >= 0) {
    float mm = __uint_as_float(maxMet[b * N_ + ns]);
    float mi = __uint_as_float(maxIou[b * N_ + ns]);
    am  = rec_m[idx] / (mm + EPS_F) * mi;
    lbl = gt_labels[b * N_ + ns];               // bg==C assumed: keep=identity
  }
  float* row = out_scores + (size_t)idx * C_;
  for (int c = 0; c < C_; c += 4) {
    v4f v = {0.0f, 0.0f, 0.0f, 0.0f};
    if (lbl >= c && lbl < c + 4) v[lbl - c] = am;
    __builtin_nontemporal_store(v, (v4f*)(row + c));
  }
}

// ---------------------------------------------------------------------------
extern "C" void kernel_launch(void* const* d_in, const int* in_sizes, int n_in,
                              void* d_out, int out_size, void* d_ws,
                              size_t ws_size, hipStream_t stream) {
  (void)in_sizes; (void)n_in; (void)out_size; (void)ws_size;

  const float* pred_scores   = (const float*)d_in[0];
  const float* pred_bboxes   = (const float*)d_in[1];
  const float* anchor_points = (const float*)d_in[2];
  /* d_in[3] = num_anchors_list (unused) */
  const int*   gt_labels     = (const int*)d_in[4];
  const float* gt_bboxes     = (const float*)d_in[5];
  const float* pad_gt        = (const float*)d_in[6];
  const int*   bg_p          = (const int*)d_in[7];

  // workspace layout (bytes)
  char* ws = (char*)d_ws;
  unsigned int* maskA  = (unsigned int*)(ws);            // B*L*2*4 = 1,075,200
  unsigned int* maxMet = (unsigned int*)(ws + 1075200);  // 4,096
  unsigned int* maxIou = (unsigned int*)(ws + 1079296);  // 4,096
  int*          rec_n  = (int*)         (ws + 1083392);  // 537,600
  float*        rec_m  = (float*)       (ws + 1620992);  // 537,600

  // output layout (floats, tuple concatenated)
  float* out_labels  = (float*)d_out;                          // B*L
  float* out_bboxes  = out_labels + (size_t)B_ * L_;           // B*L*4
  float* out_scores  = out_bboxes + (size_t)B_ * L_ * 4;       // B*L*C
  float* out_maskpos = out_scores + (size_t)B_ * L_ * C_;      // B*N*L

  (void)hipMemsetAsync(ws, 0, 1083392, stream);   // maskA + maxMet + maxIou

  TAA_topk<<<dim3(B_ * N_), dim3(256), 0, stream>>>(
      pred_scores, pred_bboxes, anchor_points, gt_labels, gt_bboxes, pad_gt,
      maskA);

  TAA_resolve<<<dim3((L_ + 255) / 256, B_), dim3(256), 0, stream>>>(
      pred_scores, pred_bboxes, gt_labels, gt_bboxes, bg_p, maskA,
      maxMet, maxIou, rec_n, rec_m, out_labels, out_bboxes, out_maskpos);

  TAA_scores<<<dim3((B_ * L_ + 255) / 256), dim3(256), 0, stream>>>(
      rec_n, rec_m, maxMet, maxIou, gt_labels, out_scores);
}